// bilateral_Filter_52888227283817
// MI455X (gfx1250) — compile-verified
//
#include <hip/hip_runtime.h>

// Bilateral 5x5, f32, ksize=5 -> pad=2, sigma_color = sigma_space = 0.15*5+0.35 = 1.1
// Both the color-weight normalization and the space-kernel normalization cancel in
// num/den, so: out = sum(exp2(d*d*C + L[ki]+L[kj]) * p) / sum(exp2(...)),
//   C    = -1/(2*1.1^2*ln2)        = -0.59615427
//   L[.] = log2(exp(-x^2/(2*1.1^2))) for x in {-2,-1,0,1,2}
//        = {-2.38461685, -0.59615427, 0, -0.59615427, -2.38461685}

#define KS   5
#define PAD  2
#define TW   64            // output tile width
#define TH   16            // output tile height
#define HTW  (TW + 2*PAD)  // 68 halo width
#define HTH  (TH + 2*PAD)  // 20 halo height
#define NTHR 256           // 8 wave32

__global__ __launch_bounds__(NTHR)
void bilateral_Filter_52888227283817_kernel(const float* __restrict__ in,
                                            float* __restrict__ out,
                                            int H, int W)
{
    __shared__ float tile[HTH * HTW];   // 5440 B of 320 KB LDS

    const int plane = blockIdx.z;                     // b*C + c
    const float* __restrict__ src = in  + (size_t)plane * H * W;
    float*       __restrict__ dst = out + (size_t)plane * H * W;
    const int tx0 = blockIdx.x * TW;
    const int ty0 = blockIdx.y * TH;
    const int tid = threadIdx.x;

    // ---- Halo tile: global -> LDS via CDNA5 async loads (ASYNCcnt path) ----
    // Reflect-101 border per element, per-lane byte addresses.
    for (int i = tid; i < HTH * HTW; i += NTHR) {
        int ly = i / HTW;
        int lx = i - ly * HTW;
        int gy = ty0 + ly - PAD;
        int gx = tx0 + lx - PAD;
        gy = (gy < 0) ? -gy : ((gy >= H) ? (2 * H - 2 - gy) : gy);
        gx = (gx < 0) ? -gx : ((gx >= W) ? (2 * W - 2 - gx) : gx);
        const float* gp = src + (size_t)gy * W + gx;
        unsigned lds_addr = (unsigned)(size_t)(&tile[i]);   // low 32 bits = LDS offset
        asm volatile("global_load_async_to_lds_b32 %0, %1, off"
                     :: "v"(lds_addr), "v"(gp)
                     : "memory");
    }
    asm volatile("s_wait_asynccnt 0" ::: "memory");  // my wave's async loads done
    __syncthreads();                                  // every wave's loads done

    // ---- Compute: 4 consecutive pixels per thread (conflict-free LDS banks) ----
    const float C  = -0.59615427f;
    const float L0 = -2.38461685f;
    const float L1 = -0.59615427f;
    const float Lt[KS] = { L0, L1, 0.0f, L1, L0 };

    const int py  = tid >> 4;          // 0..15
    const int px0 = (tid & 15) * 4;    // 0,4,...,60

    float4 res;
    float* rp = &res.x;
    #pragma unroll
    for (int s = 0; s < 4; ++s) {
        const int px = px0 + s;
        const float c = tile[(py + PAD) * HTW + (px + PAD)];
        float num = 0.0f, den = 0.0f;
        #pragma unroll
        for (int ki = 0; ki < KS; ++ki) {
            #pragma unroll
            for (int kj = 0; kj < KS; ++kj) {
                const float p = tile[(py + ki) * HTW + (px + kj)];
                const float d = p - c;
                const float a = fmaf(d * d, C, Lt[ki] + Lt[kj]); // space wt folded into exponent
                const float w = __builtin_amdgcn_exp2f(a);       // raw v_exp_f32
                num = fmaf(w, p, num);
                den += w;
            }
        }
        rp[s] = num / den;
    }
    // 16B-aligned vector store -> global_store_b128
    *(float4*)(dst + (size_t)(ty0 + py) * W + (tx0 + px0)) = res;
}

extern "C" void kernel_launch(void* const* d_in, const int* in_sizes, int n_in,
                              void* d_out, int out_size, void* d_ws, size_t ws_size,
                              hipStream_t stream) {
    (void)n_in; (void)d_ws; (void)ws_size; (void)out_size;
    const float* x = (const float*)d_in[0];
    float* y = (float*)d_out;
    const int H = 512, W = 512;
    const int planes = in_sizes[0] / (H * W);         // 16*3 = 48
    dim3 grid(W / TW, H / TH, planes);                // 8 x 32 x 48
    bilateral_Filter_52888227283817_kernel<<<grid, NTHR, 0, stream>>>(x, y, H, W);
}